// DependencyParseModel_56745107915244
// MI455X (gfx1250) — compile-verified
//
#include <hip/hip_runtime.h>
#include <hip/hip_bf16.h>

// ---------------------------------------------------------------------------
// DependencyParseModel on gfx1250 (MI455X), fp32 via V_WMMA_F32_16X16X4_F32.
//
// Sizes: SEQ=384, D=H=128, BI=256, PAIR=512, NLABELS=40, out=385*385+384*40.
// Arc scorer factored: hid(i,j)=tanh(U[i]+V[j]) with U=hV@W1[:, :256]^T + b1,
// V=hV@W1[:,256:]^T  -> avoids the 77 GFLOP pairwise GEMM.
// ---------------------------------------------------------------------------

typedef __attribute__((ext_vector_type(2))) float v2f;
typedef __attribute__((ext_vector_type(8))) float v8f;

#define SEQ 384
#define Hdim 128
#define BI 256
#define PAIR 512
#define NLAB 40
#define NP1 385   // SEQ+1

// ---------------------------------------------------------------------------
// zero-fill output buffer (score borders/diagonal + masked label rows stay 0)
// ---------------------------------------------------------------------------
__global__ void dp_zero_kernel(float* __restrict__ p, int n) {
  int i = blockIdx.x * blockDim.x + threadIdx.x;
  if (i < n) p[i] = 0.0f;
}

// ---------------------------------------------------------------------------
// embedding gather: x[t] = [word_emb[words[t]] (100) | tag_emb[tags[t]] (28)]
// ---------------------------------------------------------------------------
__global__ void dp_embed_kernel(const int* __restrict__ words,
                                const int* __restrict__ tags,
                                const float* __restrict__ wemb,
                                const float* __restrict__ temb,
                                float* __restrict__ x) {
  int t = blockIdx.x;
  int c = threadIdx.x;            // 128 threads
  float v;
  if (c < 100) v = wemb[words[t] * 100 + c];
  else         v = temb[tags[t] * 28 + (c - 100)];
  x[t * 128 + c] = v;
}

// ---------------------------------------------------------------------------
// fp32 WMMA GEMM:  C[M,N] = A[M,K] @ B^T  (B stored [N][K], row stride ldb)
//                  + bias1[n] + bias2[n], optional tanh epilogue.
// One wave -> one 16x16 tile, V_WMMA_F32_16X16X4_F32 over K in steps of 4.
// A 16x4 layout (ISA 7.12.2): lanes 0-15 hold K={0,1}, lanes 16-31 K={2,3}.
// B  4x16 layout mirrors A; D: VGPR r -> (M=r, N=lane) / (M=r+8, N=lane-16).
// M,N multiples of 16; K multiple of 4. EXEC uniform per wave.
// ---------------------------------------------------------------------------
__global__ void dp_gemm_wmma_kernel(const float* __restrict__ A, int lda,
                                    const float* __restrict__ B, int ldb,
                                    const float* __restrict__ bias1,
                                    const float* __restrict__ bias2,
                                    float* __restrict__ C, int ldc,
                                    int M, int N, int K, int do_tanh) {
  const int wave = (blockIdx.x * blockDim.x + threadIdx.x) >> 5;
  const int lane = threadIdx.x & 31;
  const int ntN = N >> 4;
  const int ntiles = (M >> 4) * ntN;
  if (wave >= ntiles) return;                 // wave-uniform branch
  const int m0 = (wave / ntN) << 4;
  const int n0 = (wave % ntN) << 4;

  const int r16  = lane & 15;                 // A row / B col within tile
  const int koff = (lane >> 4) << 1;          // 0 for lanes 0-15, 2 for 16-31

  float bv = 0.0f;
  if (bias1) bv += bias1[n0 + r16];
  if (bias2) bv += bias2[n0 + r16];
  v8f c;
  #pragma unroll
  for (int r = 0; r < 8; ++r) c[r] = bv;

  const float* __restrict__ Ar = A + (size_t)(m0 + r16) * lda + koff;
  const float* __restrict__ Br = B + (size_t)(n0 + r16) * ldb + koff;

  for (int k = 0; k < K; k += 4) {
    v2f a, b;
    a.x = Ar[k];  a.y = Ar[k + 1];
    b.x = Br[k];  b.y = Br[k + 1];
    c = __builtin_amdgcn_wmma_f32_16x16x4_f32(false, a, false, b,
                                              (short)0, c, false, false);
  }

  const int mb = m0 + ((lane >> 4) << 3);     // +8 rows for upper half-wave
  const int nn = n0 + r16;
  #pragma unroll
  for (int r = 0; r < 8; ++r) {
    float v = c[r];
    if (do_tanh) v = tanhf(v);
    C[(size_t)(mb + r) * ldc + nn] = v;
  }
}

// ---------------------------------------------------------------------------
// BiLSTM scan for one layer (blockIdx.x: 0=forward, 1=backward), 512 threads.
// G[t][512] = x_t @ Wih^T + bih + bhh precomputed by the WMMA GEMM.
// Gate n: g = G[t][n] + dot(Whh[n,:], h).  Order i,f,g,o (jnp.split).
// ---------------------------------------------------------------------------
__global__ void dp_lstm_layer_kernel(const float* __restrict__ Gf,
                                     const float* __restrict__ Gb,
                                     const float* __restrict__ Wf,
                                     const float* __restrict__ Wb,
                                     const float* __restrict__ h0,
                                     const float* __restrict__ c0,
                                     int rowf, int rowb,
                                     float* __restrict__ out, int ldout) {
  __shared__ float h_sh[Hdim];
  __shared__ float g_sh[4 * Hdim];
  const int dir = blockIdx.x;
  const float* __restrict__ G = dir ? Gb : Gf;
  const float* __restrict__ W = dir ? Wb : Wf;
  const int hrow = dir ? rowb : rowf;
  const int tid = threadIdx.x;                // 512
  float c_reg = 0.0f;
  if (tid < Hdim) {
    h_sh[tid] = h0[hrow * Hdim + tid];
    c_reg     = c0[hrow * Hdim + tid];
  }
  __syncthreads();
  float* __restrict__ ocol = out + dir * Hdim;
  const float* __restrict__ wrow = W + (size_t)tid * Hdim;

  for (int s = 0; s < SEQ; ++s) {
    const int t = dir ? (SEQ - 1 - s) : s;
    float acc = G[(size_t)t * 512 + tid];
    #pragma unroll 8
    for (int k = 0; k < Hdim; ++k) acc += wrow[k] * h_sh[k];
    g_sh[tid] = acc;
    __syncthreads();
    if (tid < Hdim) {
      float ig = g_sh[tid];
      float fg = g_sh[Hdim + tid];
      float gg = g_sh[2 * Hdim + tid];
      float og = g_sh[3 * Hdim + tid];
      float si = 1.0f / (1.0f + __expf(-ig));
      float sf = 1.0f / (1.0f + __expf(-fg));
      float so = 1.0f / (1.0f + __expf(-og));
      c_reg = sf * c_reg + si * tanhf(gg);
      float h = so * tanhf(c_reg);
      h_sh[tid] = h;
      ocol[(size_t)t * ldout + tid] = h;
    }
    __syncthreads();
  }
}

// ---------------------------------------------------------------------------
// arc scores: score[i+1][j+1] = (i!=j) * (b2 + sum_h w2[h]*tanh(U[i][h]+V[j][h]))
// 16x16 pair tile per block (256 threads); U/V/w2 staged via LDS in 128-chunks.
// ---------------------------------------------------------------------------
__global__ void dp_arc_score_kernel(const float* __restrict__ U,
                                    const float* __restrict__ V,
                                    const float* __restrict__ w2,
                                    const float* __restrict__ b2,
                                    float* __restrict__ score) {
  __shared__ float Us[16 * 128];
  __shared__ float Vs[16 * 128];
  __shared__ float w2s[128];
  const int i0 = blockIdx.y << 4;
  const int j0 = blockIdx.x << 4;
  const int tid = threadIdx.x;                // 256
  const int li = tid >> 4, lj = tid & 15;
  float s = 0.0f;
  for (int hc = 0; hc < PAIR; hc += 128) {
    for (int e = tid; e < 16 * 128; e += 256) {
      int r = e >> 7, cc = e & 127;
      Us[e] = U[(size_t)(i0 + r) * PAIR + hc + cc];
      Vs[e] = V[(size_t)(j0 + r) * PAIR + hc + cc];
    }
    if (tid < 128) w2s[tid] = w2[hc + tid];
    __syncthreads();
    const float* ur = Us + li * 128;
    const float* vr = Vs + lj * 128;
    #pragma unroll 4
    for (int h = 0; h < 128; ++h) s += w2s[h] * tanhf(ur[h] + vr[h]);
    __syncthreads();
  }
  s += b2[0];
  const int i = i0 + li, j = j0 + lj;
  if (i != j) score[(size_t)(i + 1) * NP1 + (j + 1)] = s;
}

// ---------------------------------------------------------------------------
// label features: lf[t] = [hV[t] | hV[clip(heads[t+1]-1, 0, 383)]]
// ---------------------------------------------------------------------------
__global__ void dp_label_feats_kernel(const int* __restrict__ heads,
                                      const float* __restrict__ hV,
                                      float* __restrict__ lf) {
  const int t = blockIdx.x;
  const int c = threadIdx.x;                  // 256
  int j = heads[1 + t] - 1;
  if (j < 0) j = 0;
  if (j > SEQ - 1) j = SEQ - 1;
  lf[(size_t)t * PAIR + c]      = hV[(size_t)t * BI + c];
  lf[(size_t)t * PAIR + BI + c] = hV[(size_t)j * BI + c];
}

// ---------------------------------------------------------------------------
// label output: lab[t][l] = (heads[t+1]!=0) ? lhid[t] . lab_w2[l] + b2[l] : 0
// ---------------------------------------------------------------------------
__global__ void dp_label_out_kernel(const int* __restrict__ heads,
                                    const float* __restrict__ lhid,
                                    const float* __restrict__ w2,
                                    const float* __restrict__ b2,
                                    float* __restrict__ lab) {
  __shared__ float row[PAIR];
  const int t = blockIdx.x;
  const int tid = threadIdx.x;                // 64
  for (int e = tid; e < PAIR; e += 64) row[e] = lhid[(size_t)t * PAIR + e];
  __syncthreads();
  const int hd = heads[1 + t];
  if (tid < NLAB && hd != 0) {
    float acc = b2[tid];
    const float* wr = w2 + (size_t)tid * PAIR;
    #pragma unroll 4
    for (int h = 0; h < PAIR; ++h) acc += wr[h] * row[h];
    lab[(size_t)t * NLAB + tid] = acc;
  }
}

// ---------------------------------------------------------------------------
// host side
// ---------------------------------------------------------------------------
extern "C" void kernel_launch(void* const* d_in, const int* in_sizes, int n_in,
                              void* d_out, int out_size, void* d_ws, size_t ws_size,
                              hipStream_t stream) {
  // input order (flattened setup_inputs dict):
  const int*   words  = (const int*)  d_in[0];
  const int*   tags   = (const int*)  d_in[1];
  const int*   heads  = (const int*)  d_in[2];
  const float* wemb   = (const float*)d_in[3];
  const float* temb   = (const float*)d_in[4];
  // lstm: l0f, l0b, l1f, l1b each {Wih, Whh, bih, bhh}
  const float* W0f_ih = (const float*)d_in[5];
  const float* W0f_hh = (const float*)d_in[6];
  const float* b0f_ih = (const float*)d_in[7];
  const float* b0f_hh = (const float*)d_in[8];
  const float* W0b_ih = (const float*)d_in[9];
  const float* W0b_hh = (const float*)d_in[10];
  const float* b0b_ih = (const float*)d_in[11];
  const float* b0b_hh = (const float*)d_in[12];
  const float* W1f_ih = (const float*)d_in[13];
  const float* W1f_hh = (const float*)d_in[14];
  const float* b1f_ih = (const float*)d_in[15];
  const float* b1f_hh = (const float*)d_in[16];
  const float* W1b_ih = (const float*)d_in[17];
  const float* W1b_hh = (const float*)d_in[18];
  const float* b1b_ih = (const float*)d_in[19];
  const float* b1b_hh = (const float*)d_in[20];
  const float* h0     = (const float*)d_in[21];
  const float* c0     = (const float*)d_in[22];
  const float* arc_w1 = (const float*)d_in[23];
  const float* arc_b1 = (const float*)d_in[24];
  const float* arc_w2 = (const float*)d_in[25];
  const float* arc_b2 = (const float*)d_in[26];
  const float* lab_w1 = (const float*)d_in[27];
  const float* lab_b1 = (const float*)d_in[28];
  const float* lab_w2 = (const float*)d_in[29];
  const float* lab_b2 = (const float*)d_in[30];
  (void)in_sizes; (void)n_in; (void)ws_size;

  float* ws = (float*)d_ws;
  // workspace layout (floats)
  float* x     = ws;                       // 384*128
  float* GA    = ws + 49152;               // 384*512 (G0f -> G1f -> U)
  float* GB    = ws + 245760;              // 384*512 (G0b -> G1b -> V)
  float* hcat0 = ws + 442368;              // 384*256
  float* hV    = ws + 540672;              // 384*256
  float* lf    = ws + 638976;              // 384*512
  float* lh    = ws + 835584;              // 384*512

  float* score = (float*)d_out;            // 385*385
  float* lab   = score + NP1 * NP1;        // 384*40

  auto gemm = [&](const float* A, int lda, const float* B, int ldb,
                  const float* bb1, const float* bb2, float* C, int ldc,
                  int M, int N, int K, int do_tanh) {
    int tiles = (M / 16) * (N / 16);
    int blocks = (tiles + 3) / 4;           // 4 waves (128 thr) per block
    dp_gemm_wmma_kernel<<<blocks, 128, 0, stream>>>(A, lda, B, ldb, bb1, bb2,
                                                    C, ldc, M, N, K, do_tanh);
  };

  // 0) zero output (score diag/borders, masked label rows)
  dp_zero_kernel<<<(out_size + 255) / 256, 256, 0, stream>>>(score, out_size);

  // 1) embeddings
  dp_embed_kernel<<<SEQ, 128, 0, stream>>>(words, tags, wemb, temb, x);

  // 2) layer-0 input projections (+ both biases), then scan
  gemm(x, 128, W0f_ih, 128, b0f_ih, b0f_hh, GA, 512, SEQ, 512, 128, 0);
  gemm(x, 128, W0b_ih, 128, b0b_ih, b0b_hh, GB, 512, SEQ, 512, 128, 0);
  dp_lstm_layer_kernel<<<2, 512, 0, stream>>>(GA, GB, W0f_hh, W0b_hh,
                                              h0, c0, 0, 1, hcat0, BI);

  // 3) layer-1 input projections (K=256), then scan -> hV
  gemm(hcat0, 256, W1f_ih, 256, b1f_ih, b1f_hh, GA, 512, SEQ, 512, 256, 0);
  gemm(hcat0, 256, W1b_ih, 256, b1b_ih, b1b_hh, GB, 512, SEQ, 512, 256, 0);
  dp_lstm_layer_kernel<<<2, 512, 0, stream>>>(GA, GB, W1f_hh, W1b_hh,
                                              h0, c0, 2, 3, hV, BI);

  // 4) arc factorization: U = hV @ W1[:, :256]^T + b1 ; V = hV @ W1[:,256:]^T
  gemm(hV, BI, arc_w1,       PAIR, arc_b1, nullptr, GA, PAIR, SEQ, PAIR, BI, 0);
  gemm(hV, BI, arc_w1 + BI,  PAIR, nullptr, nullptr, GB, PAIR, SEQ, PAIR, BI, 0);

  // 5) pairwise scores
  dp_arc_score_kernel<<<dim3(SEQ / 16, SEQ / 16), 256, 0, stream>>>(
      GA, GB, arc_w2, arc_b2, score);

  // 6) labels
  dp_label_feats_kernel<<<SEQ, 256, 0, stream>>>(heads, hV, lf);
  gemm(lf, PAIR, lab_w1, PAIR, lab_b1, nullptr, lh, PAIR, SEQ, PAIR, PAIR, 1);
  dp_label_out_kernel<<<SEQ, 64, 0, stream>>>(heads, lh, lab_w2, lab_b2, lab);
}